// MultiscaleTriplaneEncoder_21809843929778
// MI455X (gfx1250) — compile-verified
//
#include <hip/hip_runtime.h>

#define TILE 256
#define NCH  64

typedef __attribute__((address_space(1))) int g_int;   // global (AS1)
typedef __attribute__((address_space(3))) int l_int;   // LDS (AS3)

// Bilinear sample with clamp folded into the floor:
//   x0 = min(floor(fx), res-2), wx = fx - x0
// identical result to reference's (x0, x1=min(x0+1,W-1), wx) formulation,
// including the fx == res-1 edge (then x0=res-2, wx=1 -> picks column res-1).
__device__ __forceinline__ float bsample(const float* __restrict__ base, int res,
                                         float cx, float cy) {
  const float mx = (float)(res - 1);
  float fx = (cx + 1.0f) * 0.5f * mx;
  float fy = (cy + 1.0f) * 0.5f * mx;
  fx = fminf(fmaxf(fx, 0.0f), mx);
  fy = fminf(fmaxf(fy, 0.0f), mx);
  int x0 = (int)fx; if (x0 > res - 2) x0 = res - 2;   // (int) == floor, fx >= 0
  int y0 = (int)fy; if (y0 > res - 2) y0 = res - 2;
  float wx = fx - (float)x0;
  float wy = fy - (float)y0;
  const float* r0 = base + (size_t)y0 * (size_t)res + (size_t)x0;
  float g00 = r0[0];
  float g01 = r0[1];
  const float* r1 = r0 + res;
  float g10 = r1[0];
  float g11 = r1[1];
  float top = fmaf(g01 - g00, wx, g00);
  float bot = fmaf(g11 - g10, wx, g10);
  return fmaf(bot - top, wy, top);
}

__global__ __launch_bounds__(TILE)
void triplane_encode_kernel(
    const float* __restrict__ pts,
    const float* __restrict__ g00, const float* __restrict__ g01, const float* __restrict__ g02,
    const float* __restrict__ g10, const float* __restrict__ g11, const float* __restrict__ g12,
    const float* __restrict__ g20, const float* __restrict__ g21, const float* __restrict__ g22,
    float* __restrict__ out, int npts)
{
  __shared__ float s_pts[TILE * 3];   // staged points (async-to-LDS)
  __shared__ float s_uv[TILE * 3];    // (uv_r, phi, theta) per point

  const int  t    = threadIdx.x;
  const int  p0   = blockIdx.x * TILE;
  const int  nloc = min(TILE, npts - p0);
  const bool full = (nloc == TILE);   // true for all but the single tail block

  // ---- Phase 0: async DMA the 3KB point tile into LDS (gfx1250 ASYNCcnt path) ----
  {
    const float* src = pts + (size_t)p0 * 3;
    if (full) {
#pragma unroll
      for (int i = 0; i < 3; ++i) {
        int idx = i * TILE + t;
        __builtin_amdgcn_global_load_async_to_lds_b32(
            (g_int*)(src + idx), (l_int*)(s_pts + idx), 0, 0);
      }
    } else {
      const int ndw = nloc * 3;
#pragma unroll
      for (int i = 0; i < 3; ++i) {
        int idx = i * TILE + t;
        if (idx < ndw) {
          __builtin_amdgcn_global_load_async_to_lds_b32(
              (g_int*)(src + idx), (l_int*)(s_pts + idx), 0, 0);
        }
      }
    }
    __builtin_amdgcn_s_wait_asynccnt(0);
  }
  __syncthreads();

  // ---- Phase 1: one thread per point; do transcendental math once per point ----
  if (t < nloc) {
    float x = s_pts[t * 3 + 0];
    float y = s_pts[t * 3 + 1];
    float z = s_pts[t * 3 + 2];
    float n = sqrtf(fmaf(x, x, fmaf(y, y, z * z)));
    n = fmaxf(n, 1e-6f);
    float inv = 1.0f / n;
    float ux = x * inv, uy = y * inv, uz = z * inv;
    float uv_r = (n - 0.75f) * (1.0f / 1.5f);
    float zc = fminf(fmaxf(uz, -1.0f + 1e-6f), 1.0f - 1e-6f);
    float theta = asinf(zc) * 0.63661977236758134308f;      // / (pi/2)
    float phi   = atan2f(uy, ux) * 0.31830988618379067154f; // / pi
    s_uv[t * 3 + 0] = uv_r;
    s_uv[t * 3 + 1] = phi;
    s_uv[t * 3 + 2] = theta;
  }
  __syncthreads();

  // ---- Phase 2: thread = (point-group, channel); channels fastest -> coalesced stores
  const int ch = t & (NCH - 1);
  int res;
  const float* b0;
  const float* b1;
  const float* b2;
  if (ch < 22) {
    res = 128; size_t o = (size_t)ch * (128u * 128u);
    b0 = g00 + o; b1 = g01 + o; b2 = g02 + o;
  } else if (ch < 43) {
    res = 256; size_t o = (size_t)(ch - 22) * (256u * 256u);
    b0 = g10 + o; b1 = g11 + o; b2 = g12 + o;
  } else {
    res = 512; size_t o = (size_t)(ch - 43) * (512u * 512u);
    b0 = g20 + o; b1 = g21 + o; b2 = g22 + o;
  }

  const int lane_grp = t >> 6;   // 4 points per iteration across 256 threads
  float* orow = out + (size_t)p0 * NCH + ch;

  if (full) {
    // branch-free hot loop: no per-iteration exec-mask games
#pragma unroll 2
    for (int i = 0; i < TILE / 4; ++i) {
      int lp = i * 4 + lane_grp;
      float u0 = s_uv[lp * 3 + 0];   // uv_r
      float u1 = s_uv[lp * 3 + 1];   // phi
      float u2 = s_uv[lp * 3 + 2];   // theta
      float v = bsample(b0, res, u0, u1)
              * bsample(b1, res, u0, u2)
              * bsample(b2, res, u1, u2);
      // streaming output: keep the 87MB grid set resident in the 192MB L2
      __builtin_nontemporal_store(v, orow + (size_t)lp * NCH);
    }
  } else {
    for (int i = 0; i < TILE / 4; ++i) {
      int lp = i * 4 + lane_grp;
      if (lp < nloc) {
        float u0 = s_uv[lp * 3 + 0];
        float u1 = s_uv[lp * 3 + 1];
        float u2 = s_uv[lp * 3 + 2];
        float v = bsample(b0, res, u0, u1)
                * bsample(b1, res, u0, u2)
                * bsample(b2, res, u1, u2);
        __builtin_nontemporal_store(v, orow + (size_t)lp * NCH);
      }
    }
  }
}

extern "C" void kernel_launch(void* const* d_in, const int* in_sizes, int n_in,
                              void* d_out, int out_size, void* d_ws, size_t ws_size,
                              hipStream_t stream)
{
  (void)n_in; (void)out_size; (void)d_ws; (void)ws_size;
  const float* pts = (const float*)d_in[0];
  const int npts = in_sizes[0] / 3;
  const float* g0 = (const float*)d_in[1];
  const float* g1 = (const float*)d_in[2];
  const float* g2 = (const float*)d_in[3];
  const float* g3 = (const float*)d_in[4];
  const float* g4 = (const float*)d_in[5];
  const float* g5 = (const float*)d_in[6];
  const float* g6 = (const float*)d_in[7];
  const float* g7 = (const float*)d_in[8];
  const float* g8 = (const float*)d_in[9];
  float* out = (float*)d_out;

  const int blocks = (npts + TILE - 1) / TILE;
  triplane_encode_kernel<<<blocks, TILE, 0, stream>>>(
      pts, g0, g1, g2, g3, g4, g5, g6, g7, g8, out, npts);
}